// SeGMA_53730040873520
// MI455X (gfx1250) — compile-verified
//
#include <hip/hip_runtime.h>
#include <hip/hip_bf16.h>
#include <math.h>

#define MROWS   4096
#define IN_DIM  64
#define CLS     10
#define ROWW    74          // batch_data row width = 64 + 10
#define LDS_S   72          // LDS row stride in floats (odd*8 -> conflict-free columns)
#define GAMMA   0.20068544f // (4/(3*4096))^0.2
#define TWO_PI  6.28318530717958647692f
#define PHI_F_A 0.21445699f // phi_f(7.5) = exp(-3.75)*9.1189458

typedef float v2f __attribute__((ext_vector_type(2)));
typedef float v8f __attribute__((ext_vector_type(8)));

// ---------------- phi() exactly per reference ----------------
__device__ __forceinline__ float phi_f_(float s) {
    float t  = s * (1.0f / 7.5f);
    float t2 = t * t;
    float poly = 1.0f + t2*(3.5156229f + t2*(3.0899424f + t2*(1.2067492f
               + t2*(0.2659732f + t2*(0.0360768f + t2*0.0045813f)))));
    return expf(-0.5f * s) * poly;
}
__device__ __forceinline__ float phi_g_(float s) {
    float u = 7.5f / s;
    float poly = 0.39894228f + u*(0.01328592f + u*(0.00225319f + u*(-0.00157565f
               + u*(0.0091628f + u*(-0.02057706f + u*(0.02635537f
               + u*(-0.01647633f + u*0.00392377f)))))));
    return sqrtf(2.0f / s) * poly;
}
__device__ __forceinline__ float phi_(float x) {
    const float a = 7.5f;
    return phi_f_(fminf(x, a)) - PHI_F_A + phi_g_(fmaxf(x, a));
}

// ------------- f32 WMMA 16x16xK layer: out = act(in @ W + bias) -------------
// in : LDS, 16 rows x K cols, stride LDS_S        W : global, K x N row-major
// ISA layouts (wave32):
//   A 16x4 : lane L (M=L%16), VGPRs = K {k0+2*(L/16), +1}
//   B 4x16 : lane L (N=L%16), VGPRs = K {k0+2*(L/16), +1}
//   C 16x16: lane L, vgpr v -> (M = v + 8*(L/16), N = L%16)
template <int K, int N, bool TANH>
__device__ __forceinline__ void wmma_layer(const float* __restrict__ in,
                                           const float* __restrict__ W,
                                           const float* __restrict__ bias,
                                           float* __restrict__ out, int lane) {
    const int hf  = lane >> 4;
    const int l16 = lane & 15;
    #pragma unroll
    for (int nt = 0; nt < N; nt += 16) {
        v8f c;
        const float bv = bias[nt + l16];      // bias depends only on N
        #pragma unroll
        for (int v = 0; v < 8; ++v) c[v] = bv;
        #pragma unroll
        for (int k0 = 0; k0 < K; k0 += 4) {
            const int ka = k0 + 2 * hf;
            v2f a, b;
            a.x = in[l16 * LDS_S + ka];
            a.y = in[l16 * LDS_S + ka + 1];
            b.x = W[ka * N + nt + l16];
            b.y = W[(ka + 1) * N + nt + l16];
            c = __builtin_amdgcn_wmma_f32_16x16x4_f32(false, a, false, b,
                                                      (short)0, c, false, false);
        }
        #pragma unroll
        for (int v = 0; v < 8; ++v) {
            float val = c[v];
            if (TANH) val = tanhf(val);
            out[(v + 8 * hf) * LDS_S + nt + l16] = val;
        }
    }
}

// ---------------- fused encoder + decoder + rec-loss partial ----------------
__global__ __launch_bounds__(32) void segma_encdec(
    const float* __restrict__ batch,
    const float* __restrict__ We1, const float* __restrict__ be1,
    const float* __restrict__ We2, const float* __restrict__ be2,
    const float* __restrict__ We3, const float* __restrict__ be3,
    const float* __restrict__ Wd1, const float* __restrict__ bd1,
    const float* __restrict__ Wd2, const float* __restrict__ bd2,
    const float* __restrict__ Wd3, const float* __restrict__ bd3,
    float* __restrict__ zout, float* __restrict__ recPartial) {
    __shared__ float sT[16 * LDS_S];   // tau tile (persists for rec loss)
    __shared__ float sA[16 * LDS_S];
    __shared__ float sB[16 * LDS_S];
    __shared__ float sZ[32];
    const int lane = threadIdx.x;
    const int r0   = blockIdx.x * 16;
    const int hf = lane >> 4, l16 = lane & 15;

    for (int idx = lane; idx < 16 * IN_DIM; idx += 32) {
        int m = idx >> 6, k = idx & 63;
        sT[m * LDS_S + k] = batch[(size_t)(r0 + m) * ROWW + k];
    }
    __syncthreads();

    wmma_layer<64, 32, true>(sT, We1, be1, sA, lane);   __syncthreads();
    wmma_layer<32, 16, true>(sA, We2, be2, sB, lane);   __syncthreads();

    // z = x2 @ We3 + be3  (16x16 @ 16x2): lane -> (row=l16, col=hf)
    {
        float zv = be3[hf];
        #pragma unroll
        for (int k = 0; k < 16; ++k) zv += sB[l16 * LDS_S + k] * We3[k * 2 + hf];
        sZ[l16 * 2 + hf] = zv;
        zout[(size_t)(r0 + l16) * 2 + hf] = zv;
    }
    __syncthreads();

    // h1 = tanh(z @ Wd1 + bd1)  (16x2 @ 2x16)
    #pragma unroll
    for (int j = 0; j < 8; ++j) {
        int idx = lane + 32 * j;
        int m = idx >> 4, n = idx & 15;
        float v = sZ[m * 2] * Wd1[n] + sZ[m * 2 + 1] * Wd1[16 + n] + bd1[n];
        sA[m * LDS_S + n] = tanhf(v);
    }
    __syncthreads();

    wmma_layer<16, 32, true>(sA, Wd2, bd2, sB, lane);   __syncthreads();

    // tau_dec = h2 @ Wd3 + bd3  (16x32 @ 32x64), fuse (tau - tau_dec)^2
    float recAcc = 0.0f;
    #pragma unroll
    for (int nt = 0; nt < 64; nt += 16) {
        v8f c;
        const float bv = bd3[nt + l16];
        #pragma unroll
        for (int v = 0; v < 8; ++v) c[v] = bv;
        #pragma unroll
        for (int k0 = 0; k0 < 32; k0 += 4) {
            const int ka = k0 + 2 * hf;
            v2f a, b;
            a.x = sB[l16 * LDS_S + ka];
            a.y = sB[l16 * LDS_S + ka + 1];
            b.x = Wd3[ka * 64 + nt + l16];
            b.y = Wd3[(ka + 1) * 64 + nt + l16];
            c = __builtin_amdgcn_wmma_f32_16x16x4_f32(false, a, false, b,
                                                      (short)0, c, false, false);
        }
        #pragma unroll
        for (int v = 0; v < 8; ++v) {
            float diff = sT[(v + 8 * hf) * LDS_S + nt + l16] - c[v];
            recAcc += diff * diff;
        }
    }
    #pragma unroll
    for (int off = 16; off; off >>= 1) recAcc += __shfl_xor(recAcc, off, 32);
    if (lane == 0) recPartial[blockIdx.x] = recAcc;
}

// ---------------- deterministic block reduce ----------------
__device__ __forceinline__ float blockReduce256(float v, float* red) {
    const int t = threadIdx.x;
    red[t] = v;
    __syncthreads();
    for (int off = 128; off; off >>= 1) {
        if (t < off) red[t] += red[t + off];
        __syncthreads();
    }
    float r = red[0];
    __syncthreads();
    return r;
}

// ---------------- class loss + CW B-term (per-row) ----------------
__global__ __launch_bounds__(256) void segma_classB(
    const float* __restrict__ batch, const float* __restrict__ z,
    const float* __restrict__ means, const float* __restrict__ variances,
    const float* __restrict__ probs,
    float* __restrict__ cePartial, float* __restrict__ denPartial,
    float* __restrict__ bPartial) {
    __shared__ float red[256];
    const int r = blockIdx.x * 256 + threadIdx.x;
    const float zx = z[r * 2], zy = z[r * 2 + 1];

    float lab[CLS], labsum = 0.0f;
    #pragma unroll
    for (int c = 0; c < CLS; ++c) {
        lab[c] = batch[(size_t)r * ROWW + IN_DIM + c];
        labsum += lab[c];
    }
    float logits[CLS], mx = -1e30f;
    #pragma unroll
    for (int c = 0; c < CLS; ++c) {   // D=2 -> -0.5*D*log(2*pi*v) = -log(2*pi*v)
        float dx = zx - means[c * 2], dy = zy - means[c * 2 + 1];
        float sq = dx * dx + dy * dy;
        float lg = logf(probs[c]) - logf(TWO_PI * variances[c]) - sq / (2.0f * variances[c]);
        logits[c] = lg;
        mx = fmaxf(mx, lg);
    }
    float se = 0.0f;
    #pragma unroll
    for (int c = 0; c < CLS; ++c) se += expf(logits[c] - mx);
    const float lse = mx + logf(se);
    float ce = 0.0f;
    #pragma unroll
    for (int c = 0; c < CLS; ++c) ce -= lab[c] * (logits[c] - lse);
    const float mask = (labsum == 1.0f) ? 1.0f : 0.0f;

    float bAcc = 0.0f;
    #pragma unroll
    for (int c = 0; c < CLS; ++c) {
        float vv = variances[c] + 2.0f * GAMMA;
        float dx = means[c * 2] - zx, dy = means[c * 2 + 1] - zy;
        float b1 = dx * dx + dy * dy;
        bAcc += 2.0f * probs[c] / (4096.0f * sqrtf(TWO_PI * vv)) * phi_(b1 / (2.0f * vv));
    }

    float ceS  = blockReduce256(ce * mask, red);
    float denS = blockReduce256(labsum, red);
    float bS   = blockReduce256(bAcc, red);
    if (threadIdx.x == 0) {
        cePartial[blockIdx.x]  = ceS;
        denPartial[blockIdx.x] = denS;
        bPartial[blockIdx.x]   = bS;
    }
}

// ---------------- CW A-term: 4096x4096 pairwise phi ----------------
__global__ __launch_bounds__(256) void segma_cwA(const float* __restrict__ z,
                                                 float* __restrict__ aPartial) {
    __shared__ float sj[256 * 2];
    __shared__ float red[256];
    const int t  = threadIdx.x;
    const int bi = blockIdx.x, bj = blockIdx.y;
    const float zix = z[(bi * 256 + t) * 2];
    const float ziy = z[(bi * 256 + t) * 2 + 1];
    sj[t * 2]     = z[(bj * 256 + t) * 2];
    sj[t * 2 + 1] = z[(bj * 256 + t) * 2 + 1];
    __syncthreads();
    const float inv4g = 1.0f / (4.0f * GAMMA);
    float acc = 0.0f;
    for (int j = 0; j < 256; ++j) {
        float dx = zix - sj[j * 2];
        float dy = ziy - sj[j * 2 + 1];
        acc += phi_((dx * dx + dy * dy) * inv4g);
    }
    float s = blockReduce256(acc, red);
    if (t == 0) aPartial[bi * 16 + bj] = s;
}

// ---------------- final combine (also CW C-term, 10x10) ----------------
__global__ __launch_bounds__(256) void segma_finalize(
    const float* __restrict__ recP, const float* __restrict__ aP,
    const float* __restrict__ ceP, const float* __restrict__ denP,
    const float* __restrict__ bP,
    const float* __restrict__ means, const float* __restrict__ variances,
    const float* __restrict__ probs, float* __restrict__ out) {
    __shared__ float red[256];
    const int t = threadIdx.x;
    float recSum = blockReduce256(recP[t], red);
    float aSum   = blockReduce256(aP[t], red);
    float ceSum  = blockReduce256(t < 16 ? ceP[t] : 0.0f, red);
    float denSum = blockReduce256(t < 16 ? denP[t] : 0.0f, red);
    float bSum   = blockReduce256(t < 16 ? bP[t] : 0.0f, red);

    float cLoc = 0.0f;
    if (t < 100) {
        int c = t / 10, d = t % 10;
        float vm = variances[c] + variances[d];
        float dx = means[c * 2] - means[d * 2];
        float dy = means[c * 2 + 1] - means[d * 2 + 1];
        float c1 = dx * dx + dy * dy;
        float c2 = phi_(c1 / (2.0f * vm + 4.0f * GAMMA));
        float c3 = probs[c] * probs[d] / sqrtf(TWO_PI * (vm + 2.0f * GAMMA));
        cLoc = c3 * c2;
    }
    float cSum = blockReduce256(cLoc, red);

    if (t == 0) {
        float rec = recSum * (1.0f / (4096.0f * 64.0f));
        float A   = aSum / (4096.0f * 4096.0f * sqrtf(TWO_PI * 2.0f * GAMMA));
        float den = (denSum == 0.0f) ? 1.0f : denSum;
        float cls = ceSum / den;
        float cw  = A - bSum + cSum;
        out[0] = rec + (8.0f + logf(cw)) + 2.0f * cls;
    }
}

extern "C" void kernel_launch(void* const* d_in, const int* in_sizes, int n_in,
                              void* d_out, int out_size, void* d_ws, size_t ws_size,
                              hipStream_t stream) {
    const float* batch = (const float*)d_in[0];
    const float* We1 = (const float*)d_in[1];  const float* be1 = (const float*)d_in[2];
    const float* We2 = (const float*)d_in[3];  const float* be2 = (const float*)d_in[4];
    const float* We3 = (const float*)d_in[5];  const float* be3 = (const float*)d_in[6];
    const float* Wd1 = (const float*)d_in[7];  const float* bd1 = (const float*)d_in[8];
    const float* Wd2 = (const float*)d_in[9];  const float* bd2 = (const float*)d_in[10];
    const float* Wd3 = (const float*)d_in[11]; const float* bd3 = (const float*)d_in[12];
    const float* means = (const float*)d_in[13];
    const float* variances = (const float*)d_in[14];
    const float* probs = (const float*)d_in[15];

    float* ws   = (float*)d_ws;
    float* z    = ws;           // 4096*2 = 8192 floats
    float* recP = ws + 8192;    // 256
    float* aP   = ws + 8448;    // 256
    float* ceP  = ws + 8704;    // 16
    float* denP = ws + 8720;    // 16
    float* bP   = ws + 8736;    // 16

    segma_encdec<<<256, 32, 0, stream>>>(batch, We1, be1, We2, be2, We3, be3,
                                         Wd1, bd1, Wd2, bd2, Wd3, bd3, z, recP);
    segma_classB<<<16, 256, 0, stream>>>(batch, z, means, variances, probs,
                                         ceP, denP, bP);
    segma_cwA<<<dim3(16, 16), 256, 0, stream>>>(z, aP);
    segma_finalize<<<1, 256, 0, stream>>>(recP, aP, ceP, denP, bP,
                                          means, variances, probs, (float*)d_out);
}